// QuantumKANRegressor_25220047962616
// MI455X (gfx1250) — compile-verified
//
#include <hip/hip_runtime.h>
#include <math.h>

// QuantumKAN: VALU-bound (≈210 MFLOP fp32 + 4.2M tanh vs only 16.8 MB traffic).
// Strategy: Chebyshev recurrences (pure FMA, no trig), per-wave 16x16 tiles in
// WMMA C-layout, final feature reduction via chained V_WMMA_F32_16X16X4_F32
// against an all-ones B matrix (A staged through LDS into A-layout).

typedef float v2f __attribute__((ext_vector_type(2)));
typedef float v8f __attribute__((ext_vector_type(8)));

#define NF   64   // features
#define NDEG 16   // QSVT Chebyshev degree (T_1..T_16)
#define NKD  6    // KAN_DEG + 1 (T_0..T_5)

__global__ __launch_bounds__(256)
void qkan_kernel(const float* __restrict__ X,     // [B, 64]
                 const float* __restrict__ W,     // [64, 16] lcu weights
                 const float* __restrict__ CK,    // [64, 6]  kan coeffs
                 float* __restrict__ out,         // [B]
                 int Brows)
{
    // One 16x16 f32 tile per wave for the cross-lane transpose into A-layout.
    __shared__ float Slds[8][16][16];

    const int tid  = threadIdx.x;
    const int wave = tid >> 5;
    const int lane = tid & 31;
    const int n    = lane & 15;   // N column of the tile (feature column)
    const int half = lane >> 4;   // lane half (C-layout: rows j vs j+8)
    const int b0   = blockIdx.x * 128 + wave * 16;   // first batch row of tile

    float acc[8];
    #pragma unroll
    for (int j = 0; j < 8; ++j) acc[j] = 0.0f;

    // ---- main pipeline: 4 passes x 8 row-slots, feature f = 16*p + n ----
    #pragma unroll
    for (int p = 0; p < 4; ++p) {
        const int f = p * 16 + n;

        // per-feature weights: tiny, L2-resident, consecutive -> b128 loads
        float wv[NDEG];
        #pragma unroll
        for (int d = 0; d < NDEG; ++d) wv[d] = W[f * NDEG + d];

        float den = 0.0f;
        #pragma unroll
        for (int d = 0; d < NDEG; ++d) den += fabsf(wv[d]);
        const float rden = 1.0f / den;

        float cv[NKD];
        #pragma unroll
        for (int k = 0; k < NKD; ++k) cv[k] = CK[f * NKD + k];

        #pragma unroll
        for (int j = 0; j < 8; ++j) {
            int row = b0 + j + 8 * half;          // C-layout row mapping
            row = (row < Brows) ? row : (Brows - 1);   // clamp, keep EXEC full
            const float x = X[(size_t)row * NF + f];

            // QSVT block: s = sum_d w[d] * T_{d+1}(x)  (pure FMA recurrence)
            float tprev = 1.0f;      // T_0
            float tcur  = x;         // T_1
            float s = wv[0] * tcur;
            #pragma unroll
            for (int d = 1; d < NDEG; ++d) {
                const float tn = __builtin_fmaf(2.0f * x, tcur, -tprev);
                s = __builtin_fmaf(wv[d], tn, s);
                tprev = tcur; tcur = tn;
            }

            // KAN block: z = tanh(feat); s2 = sum_k c[k] * T_k(z)
            const float z  = tanhf(s * rden);
            float u0 = 1.0f, u1 = z;
            float s2 = __builtin_fmaf(cv[1], z, cv[0]);
            #pragma unroll
            for (int k = 2; k < NKD; ++k) {
                const float un = __builtin_fmaf(2.0f * z, u1, -u0);
                s2 = __builtin_fmaf(cv[k], un, s2);
                u0 = u1; u1 = un;
            }
            acc[j] += s2;     // S[m][n] partial (f == n mod 16 group)
        }
    }

    // ---- stage S (C-layout in registers) into LDS: S[m][n] ----
    #pragma unroll
    for (int j = 0; j < 8; ++j)
        Slds[wave][j + 8 * half][n] = acc[j];
    __syncthreads();   // cheap; guarantees DS ordering before A-layout reads

    // ---- row-sum R[m] = sum_n S[m][n] via chained WMMA f32 16x16x4 ----
    // A-layout (32-bit 16x4): lanes 0-15 hold M=lane, K={0,1}; lanes 16-31 K={2,3}.
    const int ma = n;                       // A-layout row = lane & 15
    const v2f bones = {1.0f, 1.0f};         // B[k][n] = 1 for all lanes/VGPRs
    v8f dacc = {};
    #pragma unroll
    for (int c = 0; c < 4; ++c) {
        v2f a;
        a.x = Slds[wave][ma][4 * c + 2 * half + 0];
        a.y = Slds[wave][ma][4 * c + 2 * half + 1];
        // (neg_a, A, neg_b, B, c_mod, C, reuse_a, reuse_b)
        dacc = __builtin_amdgcn_wmma_f32_16x16x4_f32(
            false, a, false, bones, (short)0, dacc, false, false);
    }

    // D C-layout: lane l, vgpr j -> R[j + 8*half] (same for every column).
    // Lane 0 writes rows b0..b0+7, lane 16 writes rows b0+8..b0+15.
    if (n == 0) {
        const int rowBase = b0 + 8 * half;
        #pragma unroll
        for (int j = 0; j < 8; ++j) {
            const int r = rowBase + j;
            if (r < Brows) out[r] = dacc[j];
        }
    }
}

extern "C" void kernel_launch(void* const* d_in, const int* in_sizes, int n_in,
                              void* d_out, int out_size, void* d_ws, size_t ws_size,
                              hipStream_t stream) {
    const float* X  = (const float*)d_in[0];   // [B, 64]
    const float* W  = (const float*)d_in[1];   // [64, 16]
    const float* CK = (const float*)d_in[2];   // [1, 64, 6]
    float* out = (float*)d_out;                // [B, 1]

    const int Brows  = in_sizes[0] / NF;       // 65536
    const int blocks = (Brows + 127) / 128;    // 128 rows per 256-thread block
    hipLaunchKernelGGL(qkan_kernel, dim3(blocks), dim3(256), 0, stream,
                       X, W, CK, out, Brows);
}